// SoftmaxGaussian_83837761618031
// MI455X (gfx1250) — compile-verified
//
#include <hip/hip_runtime.h>
#include <stdint.h>

// ---------------------------------------------------------------------------
// Monte-Carlo softmax-of-Gaussian: out_mean/out_var over 10000 samples.
// BATCH=512 rows, K=16 categories, S=10000 samples.
// One block (8 wave32) per row; each wave owns a 1250-sample slice.
// Per wave-iteration: a 16(k) x 16(sample) fp32 tile of exp() values is
// reduced over k with 4 chained V_WMMA_F32_16X16X4_F32 (A = ones).
// Hot-loop transcendentals use raw v_log/v_sqrt/v_cos/v_exp (inputs are
// guaranteed normal & in-range, so the libm guard code is dead weight).
// ---------------------------------------------------------------------------

typedef __attribute__((ext_vector_type(2))) float v2f;
typedef __attribute__((ext_vector_type(8))) float v8f;

#define BATCH 512
#define KCAT  16
#define NSAMP 10000
#define SLICES 8                           // waves per row
#define SAMP_PER_SLICE (NSAMP / SLICES)    // 1250
#define FULL_TILES (SAMP_PER_SLICE / 16)   // 78 unconditional tiles
// epilogue tile covers the remaining SAMP_PER_SLICE % 16 = 2 samples

#define NEG_TWO_LN2  (-1.3862943611198906f)  // -2*ln(2): -2*ln(u) = this * log2(u)
#define LOG2E        ( 1.4426950408889634f)  // exp(x) = exp2(x * LOG2E)

__device__ __forceinline__ uint32_t hash32(uint32_t x) {
  x ^= x >> 16; x *= 0x7feb352dU;
  x ^= x >> 15; x *= 0x846ca68bU;
  x ^= x >> 16;
  return x;
}

// Deterministic N(0,1) draw keyed by a 32-bit counter (Box-Muller, cosine leg).
// u1 in (2^-33, 1]: always a normal float -> raw v_log_f32 is safe & accurate.
// v_cos_f32 computes cos(2*pi*x) natively, so u2 is passed in revolutions.
__device__ __forceinline__ float normal_from_counter(uint32_t idx) {
  uint32_t h1 = hash32(idx ^ 0x9E3779B9u);
  uint32_t h2 = hash32(idx + 0x85EBCA6Bu);
  float u1 = (float)(h1 | 1u) * 2.3283064365386963e-10f;  // (0,1]
  float u2 = (float)h2 * 2.3283064365386963e-10f;         // [0,1) revolutions
  float l2 = __builtin_amdgcn_logf(u1);                   // log2(u1), one v_log
  float r  = __builtin_amdgcn_sqrtf(NEG_TWO_LN2 * l2);    // sqrt(-2 ln u1)
  return r * __builtin_amdgcn_cosf(u2);                   // * cos(2*pi*u2)
}

// One 16x16 (category x sample) tile: generate exps, WMMA column-sum,
// normalize, accumulate sum / sum-of-squares. FULL=true => no predication.
template <bool FULL>
__device__ __forceinline__ void
do_tile(int t, int b, int hi, int n, int s_begin,
        const float (&mu)[4][2], const float (&sd)[4][2],
        float (&acc)[4][2], float (&acq)[4][2], const v2f ones) {
  const int scol = t * 16 + n;              // sample offset in slice
  const int s    = s_begin + scol;          // sample index in row
  const bool valid = FULL || (scol < SAMP_PER_SLICE);

  float ex[4][2];
  v2f   e[4];
#pragma unroll
  for (int c = 0; c < 4; ++c) {
#pragma unroll
    for (int j = 0; j < 2; ++j) {
      const int k = 4 * c + 2 * j + hi;
      const uint32_t idx = (uint32_t)((b * KCAT + k) * NSAMP + s);
      const float z = normal_from_counter(idx);
      // exp(z*sd+mu) = exp2((z*sd+mu)*log2e); |arg| < ~16 so raw v_exp is fine
      ex[c][j] = __builtin_amdgcn_exp2f((z * sd[c][j] + mu[c][j]) * LOG2E);
    }
    e[c][0] = ex[c][0];
    e[c][1] = ex[c][1];
  }

  // Column sums via WMMA: D[m][n] = sum_k E[k][n] for every m.
  v8f csum = {};
#pragma unroll
  for (int c = 0; c < 4; ++c) {
    csum = __builtin_amdgcn_wmma_f32_16x16x4_f32(
        /*neg_a=*/false, ones, /*neg_b=*/false, e[c],
        /*c_mod=*/(short)0, csum, /*reuse_a=*/false, /*reuse_b=*/false);
  }
  // Fast reciprocal (v_rcp_f32): ~1 ulp, far below the ~1% MC noise floor.
  const float inv = __builtin_amdgcn_rcpf(csum[0]);

  if (valid) {
#pragma unroll
    for (int c = 0; c < 4; ++c)
#pragma unroll
      for (int j = 0; j < 2; ++j) {
        const float p = ex[c][j] * inv;
        acc[c][j] += p;
        acq[c][j] += p * p;
      }
  }
}

__global__ void __launch_bounds__(256)
mc_softmax_accum_kernel(const float* __restrict__ mean,
                        const float* __restrict__ var,
                        float* __restrict__ ws_sum,   // [BATCH*KCAT] sum of p
                        float* __restrict__ ws_sq) {  // [BATCH*KCAT] sum of p^2
  const int b    = blockIdx.x;            // one row per block
  const int lane = threadIdx.x & 31;
  const int wave = threadIdx.x >> 5;      // slice id 0..7
  const int n    = lane & 15;             // sample column within tile
  const int hi   = lane >> 4;             // half-wave selects odd/even k

  // Each lane owns 8 categories: k = 4*c + 2*j + hi  (c=0..3 chunk, j=0..1 vgpr)
  float mu[4][2], sd[4][2];
#pragma unroll
  for (int c = 0; c < 4; ++c)
#pragma unroll
    for (int j = 0; j < 2; ++j) {
      const int k = 4 * c + 2 * j + hi;
      mu[c][j] = mean[b * KCAT + k];
      sd[c][j] = __builtin_amdgcn_sqrtf(var[b * KCAT + k]);
    }

  float acc[4][2] = {{0.f, 0.f}, {0.f, 0.f}, {0.f, 0.f}, {0.f, 0.f}};
  float acq[4][2] = {{0.f, 0.f}, {0.f, 0.f}, {0.f, 0.f}, {0.f, 0.f}};

  const int s_begin = wave * SAMP_PER_SLICE;
  const v2f ones = {1.0f, 1.0f};          // A matrix (16x4) of all ones

  // Steady state: 78 full tiles, no exec-mask manipulation.
  for (int t = 0; t < FULL_TILES; ++t)
    do_tile<true>(t, b, hi, n, s_begin, mu, sd, acc, acq, ones);
  // Epilogue: last partial tile (2 valid sample columns).
  do_tile<false>(FULL_TILES, b, hi, n, s_begin, mu, sd, acc, acq, ones);

  // Reduce over the 16 sample columns (stays within each half-wave).
#pragma unroll
  for (int off = 1; off <= 8; off <<= 1) {
#pragma unroll
    for (int c = 0; c < 4; ++c)
#pragma unroll
      for (int j = 0; j < 2; ++j) {
        acc[c][j] += __shfl_xor(acc[c][j], off, 32);
        acq[c][j] += __shfl_xor(acq[c][j], off, 32);
      }
  }

  if (n == 0) {  // lanes 0 and 16 hold slice totals for their 8 categories
#pragma unroll
    for (int c = 0; c < 4; ++c)
#pragma unroll
      for (int j = 0; j < 2; ++j) {
        const int k = 4 * c + 2 * j + hi;
        atomicAdd(&ws_sum[b * KCAT + k], acc[c][j]);
        atomicAdd(&ws_sq[b * KCAT + k], acq[c][j]);
      }
  }
}

__global__ void __launch_bounds__(256)
mc_softmax_finalize_kernel(const float* __restrict__ ws_sum,
                           const float* __restrict__ ws_sq,
                           float* __restrict__ out) {
  const int i = blockIdx.x * blockDim.x + threadIdx.x;
  if (i < BATCH * KCAT) {
    const float s = ws_sum[i];
    const float q = ws_sq[i];
    const float m = s * (1.0f / (float)NSAMP);
    // unbiased variance (ddof=1)
    const float v = (q - (float)NSAMP * m * m) * (1.0f / (float)(NSAMP - 1));
    out[i] = m;
    out[BATCH * KCAT + i] = v;
  }
}

extern "C" void kernel_launch(void* const* d_in, const int* in_sizes, int n_in,
                              void* d_out, int out_size, void* d_ws, size_t ws_size,
                              hipStream_t stream) {
  (void)in_sizes; (void)n_in; (void)out_size; (void)ws_size;
  const float* mean = (const float*)d_in[0];
  const float* var  = (const float*)d_in[1];
  float* out = (float*)d_out;

  float* ws_sum = (float*)d_ws;
  float* ws_sq  = ws_sum + BATCH * KCAT;

  // Zero accumulators every call (graph-capture safe).
  hipMemsetAsync(d_ws, 0, 2 * BATCH * KCAT * sizeof(float), stream);

  mc_softmax_accum_kernel<<<BATCH, 256, 0, stream>>>(mean, var, ws_sum, ws_sq);

  const int nfin = BATCH * KCAT;
  mc_softmax_finalize_kernel<<<(nfin + 255) / 256, 256, 0, stream>>>(ws_sum, ws_sq, out);
}